// ChebNet_70050916598068
// MI455X (gfx1250) — compile-verified
//
#include <hip/hip_runtime.h>
#include <math.h>

typedef __attribute__((ext_vector_type(2))) float v2f;
typedef __attribute__((ext_vector_type(8))) float v8f;

// ---------------------------------------------------------------------------
// ChebNet K=2, channels 1 -> 16 -> 1.
// Key identity: (L_hat @ X) @ W == L_hat @ (X @ W), and W maps to/from 1
// channel on both propagation paths, so both graph propagations operate on a
// single scalar per node. Edge weight factorization:
//   w[e] = -dinv[row]*dinv[col]  =>  T1[i] = -dinv[i]*sum_{row=i} dinv[c]*x[c]
// ---------------------------------------------------------------------------

__global__ void k_zero2(float* __restrict__ a, float* __restrict__ b, int n) {
    int i = blockIdx.x * blockDim.x + threadIdx.x;
    if (i < n) { a[i] = 0.0f; b[i] = 0.0f; }
}

__global__ void k_zero(float* __restrict__ a, int n) {
    int i = blockIdx.x * blockDim.x + threadIdx.x;
    if (i < n) a[i] = 0.0f;
}

// deg[row[e]] += 1  (row half of edge_index)
__global__ void k_degree(const int* __restrict__ ei, float* __restrict__ deg, int e) {
    int i = blockIdx.x * blockDim.x + threadIdx.x;
    if (i < e) atomicAdd(&deg[ei[i]], 1.0f);
}

// dinv = deg>0 ? 1/sqrt(deg) : 0 (in place over deg); g = dinv * x
__global__ void k_prep(const float* __restrict__ x, float* __restrict__ degdinv,
                       float* __restrict__ g, int n) {
    int i = blockIdx.x * blockDim.x + threadIdx.x;
    if (i < n) {
        float d  = degdinv[i];
        float dv = (d > 0.0f) ? (1.0f / sqrtf(d)) : 0.0f;
        degdinv[i] = dv;
        g[i] = dv * x[i];
    }
}

// acc[row[e]] += g[col[e]]   (scalar gather + scatter-add; L2-resident targets)
__global__ void k_scatter(const int* __restrict__ ei, const float* __restrict__ g,
                          float* __restrict__ acc, int e) {
    int i = blockIdx.x * blockDim.x + threadIdx.x;
    if (i < e) atomicAdd(&acc[ei[i]], g[e + i]);
}

// Per-node dense update, 16 nodes per wave via V_WMMA_F32_16X16X4_F32:
//   A (16x4) rows = [x_m, T1_m, 0, 0],  B (4x16) rows = [W1[0,0,:]; W1[1,0,:]; 0; 0]
//   D = A@B  ->  h = relu(D + b1)
//   p_m = h . W2[0,:,0]   (written to out, partial)
//   z_m = dinv_m * (h . W2[1,:,0])  (scatter source of layer 2, written to g)
__global__ void node_update_wmma(const float* __restrict__ x,
                                 const float* __restrict__ dinv,
                                 const float* __restrict__ acc,
                                 const float* __restrict__ W1,   // (2,1,16) flat
                                 const float* __restrict__ b1,   // (16)
                                 const float* __restrict__ W2,   // (2,16,1) flat
                                 float* __restrict__ outp,       // d_out: partial p
                                 float* __restrict__ g,          // z output
                                 int n) {
    const int lane   = threadIdx.x & 31;
    const int laneLo = lane & 15;
    const bool hiHalf = lane >= 16;
    const int wave = (blockIdx.x * blockDim.x + threadIdx.x) >> 5;
    const int base = wave << 4;
    if (base >= n) return;               // wave-uniform: EXEC stays all-ones

    // --- A matrix 16x4 f32: VGPR0 = K(0|2), VGPR1 = K(1|3); M = lane&15 ---
    int m = base + laneLo;
    if (m >= n) m = n - 1;               // clamped read (N%16==0 in practice)
    float xv = x[m];
    float tv = -dinv[m] * acc[m];        // T1 = L_hat @ x  (factored form)
    v2f a;
    a.x = hiHalf ? 0.0f : xv;            // K=0 (lo) / K=2 (hi, zero pad)
    a.y = hiHalf ? 0.0f : tv;            // K=1 (lo) / K=3 (hi, zero pad)

    // --- B matrix 4x16 f32: N = lane&15, VGPR0 = K(0|2), VGPR1 = K(1|3) ---
    v2f b;
    b.x = hiHalf ? 0.0f : W1[laneLo];        // row K=0: W1[0,0,:]
    b.y = hiHalf ? 0.0f : W1[16 + laneLo];   // row K=1: W1[1,0,:]

    v8f c = {};
    // (neg_a, A, neg_b, B, c_mod, C, reuse_a, reuse_b)
    c = __builtin_amdgcn_wmma_f32_16x16x4_f32(false, a, false, b,
                                              (short)0, c, false, false);

    // D layout: vgpr r, lanes 0-15 -> node base+r, channel=lane;
    //           lanes 16-31 -> node base+r+8, channel=lane-16.
    const float bias = b1[laneLo];
    const float w20  = W2[laneLo];        // W2[0,:,0]
    const float w21  = W2[16 + laneLo];   // W2[1,:,0]

    #pragma unroll
    for (int r = 0; r < 8; ++r) {
        float h = c[r] + bias;
        h = h > 0.0f ? h : 0.0f;          // relu
        float pu = h * w20;
        float su = h * w21;
        // half-wave (16-lane) reduction over channels; xor masks 1..8 never
        // cross the 16-lane boundary at width 32.
        #pragma unroll
        for (int mask = 1; mask <= 8; mask <<= 1) {
            pu += __shfl_xor(pu, mask, 32);
            su += __shfl_xor(su, mask, 32);
        }
        if (laneLo == 0) {
            int mm = base + r + (hiHalf ? 8 : 0);
            if (mm < n) {
                outp[mm] = pu;                 // h @ W2[0]
                g[mm]    = dinv[mm] * su;      // z = dinv * (h @ W2[1])
            }
        }
    }
}

// out[i] = p[i] - dinv[i]*acc[i] + b2
__global__ void k_final(const float* __restrict__ dinv, const float* __restrict__ acc,
                        const float* __restrict__ b2, float* __restrict__ out, int n) {
    int i = blockIdx.x * blockDim.x + threadIdx.x;
    if (i < n) out[i] = out[i] - dinv[i] * acc[i] + b2[0];
}

extern "C" void kernel_launch(void* const* d_in, const int* in_sizes, int n_in,
                              void* d_out, int out_size, void* d_ws, size_t ws_size,
                              hipStream_t stream) {
    const float* x  = (const float*)d_in[0];
    const int*   ei = (const int*)d_in[1];   // (2,E): rows [0,E), cols [E,2E)
    const float* W1 = (const float*)d_in[2]; // (2,1,16)
    const float* b1 = (const float*)d_in[3]; // (16)
    const float* W2 = (const float*)d_in[4]; // (2,16,1)
    const float* b2 = (const float*)d_in[5]; // (1)
    const int n = in_sizes[0];               // N (x is (N,1))
    const int e = in_sizes[1] / 2;           // E
    float* out = (float*)d_out;

    // Workspace: 3 * N floats (deg->dinv, gather source, accumulator)
    float* degdinv = (float*)d_ws;
    float* g       = degdinv + n;
    float* acc     = g + n;

    const int BT = 256;
    const int gn = (n + BT - 1) / BT;
    const int ge = (e + BT - 1) / BT;
    const int waves = (n + 15) / 16;
    const int gw = (waves * 32 + BT - 1) / BT;

    k_zero2          <<<gn, BT, 0, stream>>>(degdinv, acc, n);
    k_degree         <<<ge, BT, 0, stream>>>(ei, degdinv, e);
    k_prep           <<<gn, BT, 0, stream>>>(x, degdinv, g, n);
    k_scatter        <<<ge, BT, 0, stream>>>(ei, g, acc, e);       // layer-1 propagate
    node_update_wmma <<<gw, BT, 0, stream>>>(x, degdinv, acc, W1, b1, W2, out, g, n);
    k_zero           <<<gn, BT, 0, stream>>>(acc, n);
    k_scatter        <<<ge, BT, 0, stream>>>(ei, g, acc, e);       // layer-2 propagate
    k_final          <<<gn, BT, 0, stream>>>(degdinv, acc, b2, out, n);
}